// QLoRALayer_20564303413746
// MI455X (gfx1250) — compile-verified
//
#include <hip/hip_runtime.h>

typedef __attribute__((ext_vector_type(2))) float v2f;
typedef __attribute__((ext_vector_type(8))) float v8f;

// bitsandbytes NF4 codebook
__device__ __constant__ float NF4_LUT[16] = {
    -1.0f, -0.6961928009986877f, -0.5250730514526367f, -0.39491748809814453f,
    -0.28444138169288635f, -0.18477343022823334f, -0.09105003625154495f, 0.0f,
    0.07958029955625534f, 0.16093020141124725f, 0.24611230194568634f,
    0.33791524171829224f, 0.44070982933044434f, 0.5626170039176941f,
    0.7229568362236328f, 1.0f};

#define B_DIM   4
#define S_DIM   2048
#define IN_DIM  4096
#define OUT_DIM 4096
#define R_DIM   8
#define NTOK    (B_DIM * S_DIM)      // 8192 tokens
#define SCALING 4.0f                 // alpha/rank = 32/8

// ---------------------------------------------------------------------------
// Dequant W_A into WMMA B-fragment order for V_WMMA_F32_16X16X4_F32.
// B matrix (4xK step s): lane l, vgpr v = W_A[n][k], n = l%16 (pad r 8..15 -> 0),
// k = 4*s + 2*(l/16) + v.  Stored as fragWA[s*64 + l*2 + v], s in [0,1024).
// ---------------------------------------------------------------------------
__global__ void dq_A_kernel(const int* __restrict__ codes_A,
                            const float* __restrict__ absmax_A,
                            float* __restrict__ fragWA) {
    int t   = blockIdx.x * 256 + threadIdx.x;   // 65536 total
    int s   = t >> 6;
    int rem = t & 63;
    int l   = rem >> 1;
    int v   = rem & 1;
    int n   = l & 15;
    int k   = 4 * s + 2 * (l >> 4) + v;
    float val = 0.0f;
    if (n < R_DIM) {
        int idx = n * IN_DIM + k;               // row-major over [R, IN]
        val = NF4_LUT[codes_A[idx] & 15] * absmax_A[idx >> 6];
    }
    fragWA[t] = val;
}

// ---------------------------------------------------------------------------
// Dequant W_B (x SCALING) into B-fragment order for GEMM2 (K = r).
// Tile ot covers outputs [ot*16, ot*16+16); step t in {0,1} covers r=0..7.
// fragB2[((ot*2 + t)*32 + l)*2 + v] = SCALING * W_B[ot*16 + l%16][4t + 2*(l/16) + v]
// ---------------------------------------------------------------------------
__global__ void dq_B_kernel(const int* __restrict__ codes_B,
                            const float* __restrict__ absmax_B,
                            float* __restrict__ fragB2) {
    int t  = blockIdx.x * 256 + threadIdx.x;    // 32768 total
    int v  = t & 1;
    int l  = (t >> 1) & 31;
    int st = (t >> 6) & 1;
    int ot = t >> 7;
    int o  = ot * 16 + (l & 15);
    int r  = 4 * st + 2 * (l >> 4) + v;         // always < 8
    int idx = o * R_DIM + r;                    // row-major over [OUT, R]
    fragB2[t] = NF4_LUT[codes_B[idx] & 15] * absmax_B[idx >> 6] * SCALING;
}

// ---------------------------------------------------------------------------
// Fused low-rank kernel: one 16-token tile per workgroup (8 waves).
//  GEMM1: interm[16x16] = x_tile[16x4096] @ fragWA      (K split over 8 waves)
//  GEMM2: out_tile[16x4096] = interm @ fragB2           (256 N-tiles / 8 waves)
// ---------------------------------------------------------------------------
__global__ __launch_bounds__(256) void qlora_main_kernel(
        const float* __restrict__ x,
        const float* __restrict__ fragWA,
        const float* __restrict__ fragB2,
        float* __restrict__ out) {
    __shared__ float sPart[8 * 256];   // per-wave partial interm tiles
    __shared__ float sInterm[256];     // reduced interm[16][16]

    const int tok0 = blockIdx.x * 16;
    const int lane = threadIdx.x & 31;
    const int wave = threadIdx.x >> 5;
    const int m    = lane & 15;        // matrix row handled by this lane (A / N col for B,C,D)
    const int koff = 2 * (lane >> 4);  // K sub-offset within a 4-wide step
    const int rsel = (lane >> 4) * 8;  // C/D: lanes 16-31 hold rows M=8..15

    // ---------------- GEMM1: K slice [wave*512, wave*512+512) ----------------
    v8f acc0 = {0,0,0,0,0,0,0,0};
    v8f acc1 = {0,0,0,0,0,0,0,0};
    v8f acc2 = {0,0,0,0,0,0,0,0};
    v8f acc3 = {0,0,0,0,0,0,0,0};

    const float* xrow = x + (size_t)(tok0 + m) * IN_DIM + koff;
    const int kbase = wave * 512;
    #pragma unroll 2
    for (int k = kbase; k < kbase + 512; k += 16) {
        v2f a0 = *(const v2f*)(xrow + k);
        v2f a1 = *(const v2f*)(xrow + k + 4);
        v2f a2 = *(const v2f*)(xrow + k + 8);
        v2f a3 = *(const v2f*)(xrow + k + 12);
        const float* bf = fragWA + (size_t)(k >> 2) * 64 + lane * 2;
        v2f b0 = *(const v2f*)(bf);
        v2f b1 = *(const v2f*)(bf + 64);
        v2f b2 = *(const v2f*)(bf + 128);
        v2f b3 = *(const v2f*)(bf + 192);
        acc0 = __builtin_amdgcn_wmma_f32_16x16x4_f32(false, a0, false, b0,
                                                     (short)0, acc0, false, false);
        acc1 = __builtin_amdgcn_wmma_f32_16x16x4_f32(false, a1, false, b1,
                                                     (short)0, acc1, false, false);
        acc2 = __builtin_amdgcn_wmma_f32_16x16x4_f32(false, a2, false, b2,
                                                     (short)0, acc2, false, false);
        acc3 = __builtin_amdgcn_wmma_f32_16x16x4_f32(false, a3, false, b3,
                                                     (short)0, acc3, false, false);
    }
    v8f acc = (acc0 + acc1) + (acc2 + acc3);

    // deposit this wave's partial tile: D[v + rsel][m] for v = 0..7
    #pragma unroll
    for (int v = 0; v < 8; ++v)
        sPart[wave * 256 + (rsel + v) * 16 + m] = acc[v];
    __syncthreads();

    // deterministic 8-way reduction: thread t owns interm element t
    {
        float s = 0.0f;
        #pragma unroll
        for (int w = 0; w < 8; ++w)
            s += sPart[w * 256 + threadIdx.x];
        sInterm[threadIdx.x] = s;
    }
    __syncthreads();

    // ---------------- GEMM2: interm[16x16] @ W_B^T (K = 8) ------------------
    // A fragments from LDS: step t -> interm[m][4t + koff .. +1]
    v2f ia0 = *(const v2f*)(&sInterm[m * 16 + koff]);
    v2f ia1 = *(const v2f*)(&sInterm[m * 16 + 4 + koff]);

    for (int ot = wave; ot < OUT_DIM / 16; ot += 8) {
        const float* bp = fragB2 + (size_t)ot * 128 + lane * 2;
        v2f b0 = *(const v2f*)(bp);
        v2f b1 = *(const v2f*)(bp + 64);
        v8f c  = {0,0,0,0,0,0,0,0};
        c = __builtin_amdgcn_wmma_f32_16x16x4_f32(false, ia0, false, b0,
                                                  (short)0, c, false, false);
        c = __builtin_amdgcn_wmma_f32_16x16x4_f32(false, ia1, false, b1,
                                                  (short)0, c, false, false);
        float* op = out + (size_t)(tok0 + rsel) * OUT_DIM + ot * 16 + m;
        #pragma unroll
        for (int v = 0; v < 8; ++v)
            op[(size_t)v * OUT_DIM] = c[v];
    }
}

extern "C" void kernel_launch(void* const* d_in, const int* in_sizes, int n_in,
                              void* d_out, int out_size, void* d_ws, size_t ws_size,
                              hipStream_t stream) {
    const float* x        = (const float*)d_in[0];
    const int*   codes_A  = (const int*)  d_in[1];
    const float* absmax_A = (const float*)d_in[2];
    const int*   codes_B  = (const int*)  d_in[3];
    const float* absmax_B = (const float*)d_in[4];
    float*       out      = (float*)d_out;

    float* fragWA = (float*)d_ws;            // 1024*64 = 65536 floats (256 KB)
    float* fragB2 = fragWA + 65536;          // 256*2*64 = 32768 floats (128 KB)

    dq_A_kernel<<<256, 256, 0, stream>>>(codes_A, absmax_A, fragWA);
    dq_B_kernel<<<128, 256, 0, stream>>>(codes_B, absmax_B, fragB2);
    qlora_main_kernel<<<NTOK / 16, 256, 0, stream>>>(x, fragWA, fragB2, out);
}